// ATSSSSD512Loss_83167746720167
// MI455X (gfx1250) — compile-verified
//
#include <hip/hip_runtime.h>
#include <math.h>

// ---------------- problem constants (from reference) ----------------
#define N_B    32
#define N_K    16
#define N_C    80
#define N_P    8525
#define N_LEV  5
#define N_CAND 9
#define F_ALPHA_ 0.25f
#define NBLK_BG 1024

typedef __attribute__((ext_vector_type(2))) float v2f;
typedef __attribute__((ext_vector_type(8))) float v8f;

__device__ __forceinline__ float logsig(float z) {
    // numerically stable log(sigmoid(z))
    return fminf(z, 0.0f) - log1pf(expf(-fabsf(z)));
}

__device__ __forceinline__ float iou_xy(const float a[4], const float b[4]) {
    float iw = fmaxf(fminf(a[2], b[2]) - fmaxf(a[0], b[0]), 0.0f);
    float ih = fmaxf(fminf(a[3], b[3]) - fmaxf(a[1], b[1]), 0.0f);
    float inter = iw * ih;
    float aa = (a[2] - a[0]) * (a[3] - a[1]);
    float ab = (b[2] - b[0]) * (b[3] - b[1]);
    return inter / (aa + ab - inter);
}

__device__ __forceinline__ float ciou_f(const float p[4], const float g[4]) {
    float w1 = p[2] - p[0], h1 = p[3] - p[1];
    float w2 = g[2] - g[0], h2 = g[3] - g[1];
    float iw = fmaxf(fminf(p[2], g[2]) - fmaxf(p[0], g[0]), 0.0f);
    float ih = fmaxf(fminf(p[3], g[3]) - fmaxf(p[1], g[1]), 0.0f);
    float inter = iw * ih;
    float uni = w1 * h1 + w2 * h2 - inter;
    float iou = inter / uni;
    float c1x = (p[0] + p[2]) * 0.5f, c1y = (p[1] + p[3]) * 0.5f;
    float c2x = (g[0] + g[2]) * 0.5f, c2y = (g[1] + g[3]) * 0.5f;
    float rho2 = (c1x - c2x) * (c1x - c2x) + (c1y - c2y) * (c1y - c2y);
    float dx = fmaxf(p[2], g[2]) - fminf(p[0], g[0]);
    float dy = fmaxf(p[3], g[3]) - fminf(p[1], g[1]);
    float cdiag = dx * dx + dy * dy;
    const float C4PI2 = 0.40528473456935109f;   // 4/pi^2
    float dv = atanf(w2 / h2) - atanf(w1 / h1);
    float v = C4PI2 * dv * dv;
    float alpha = v / (1.0f - iou + v);
    float r = iou - rho2 / cdiag - alpha * v;
    return fminf(fmaxf(r, -1.0f), 1.0f);
}

// ------------------------------------------------------------------
// Kernel A: per-image ATSS assignment + CIoU partials + focal corrections.
// One block (16 wave32s) per image. Distances via V_WMMA_F32_16X16X4_F32:
//   A[k][*] = [-2gx_k, -2gy_k, 1, gx_k^2+gy_k^2]
//   B[*][n] = [px_n, py_n, px_n^2+py_n^2, 1]
//   D[k][n] = ||g_k - p_n||^2
// ------------------------------------------------------------------
__global__ void __launch_bounds__(512)
atss_assign_kernel(const float* __restrict__ locs,
                   const float* __restrict__ scores,
                   const float* __restrict__ boxes,
                   const int*   __restrict__ labels,
                   const float* __restrict__ priors,
                   float* __restrict__ img_part)
{
    const int SPL[6] = {0, 6400, 8000, 8400, 8500, 8525};
    const int b    = blockIdx.x;
    const int tid  = threadIdx.x;
    const int wave = tid >> 5;
    const int lane = tid & 31;

    __shared__ float s_box[N_K][4];
    __shared__ float s_gc[N_K][2];
    __shared__ int   s_lab[N_K];
    __shared__ float s_dist[N_K][256];
    __shared__ float s_mval[N_K][N_CAND * 32];
    __shared__ int   s_midx[N_K][N_CAND * 32];
    __shared__ float s_posov[N_K][N_LEV * N_CAND];
    __shared__ int   s_tidx[N_K][N_LEV * N_CAND];
    __shared__ float s_thr[N_K];
    __shared__ float s_acc[N_LEV * N_CAND][4];

    if (tid < N_K) {
        float x0 = boxes[(b * N_K + tid) * 4 + 0];
        float y0 = boxes[(b * N_K + tid) * 4 + 1];
        float x1 = boxes[(b * N_K + tid) * 4 + 2];
        float y1 = boxes[(b * N_K + tid) * 4 + 3];
        s_box[tid][0] = x0; s_box[tid][1] = y0; s_box[tid][2] = x1; s_box[tid][3] = y1;
        s_gc[tid][0] = (x0 + x1) * 0.5f;
        s_gc[tid][1] = (y0 + y1) * 0.5f;
        s_lab[tid] = labels[b * N_K + tid];
    }
    __syncthreads();

    // A fragment (per-lane): lanes 0-15 hold K=0,1 ; lanes 16-31 hold K=2,3
    const int mrow = lane & 15;
    const float gx = s_gc[mrow][0], gy = s_gc[mrow][1];
    v2f afrag;
    if (lane < 16) { afrag.x = -2.0f * gx; afrag.y = -2.0f * gy; }
    else           { afrag.x = 1.0f;       afrag.y = gx * gx + gy * gy; }

    for (int l = 0; l < N_LEV; ++l) {
        const int base = SPL[l];
        const int Pl   = SPL[l + 1] - base;

        float tv[N_CAND];
        int   tix[N_CAND];
#pragma unroll
        for (int j = 0; j < N_CAND; ++j) { tv[j] = 3.0e38f; tix[j] = 0; }

        const int nblk = (Pl + 255) >> 8;
        for (int blk = 0; blk < nblk; ++blk) {
            // ---- build B fragment for this wave's 16-prior chunk ----
            int ploc = blk * 256 + wave * 16 + (lane & 15);
            int pc   = min(ploc, Pl - 1);
            float px = priors[(base + pc) * 4 + 0];
            float py = priors[(base + pc) * 4 + 1];
            v2f bfrag;
            if (lane < 16) { bfrag.x = px;                bfrag.y = py;   }
            else           { bfrag.x = px * px + py * py; bfrag.y = 1.0f; }

            v8f cfrag = {};
            v8f d = __builtin_amdgcn_wmma_f32_16x16x4_f32(
                        false, afrag, false, bfrag, (short)0, cfrag, false, false);

            // ---- scatter 16x16 tile to LDS: lanes 0-15 -> rows 0-7, 16-31 -> rows 8-15
            int col = wave * 16 + (lane & 15);
            int rb  = (lane < 16) ? 0 : 8;
            s_dist[rb + 0][col] = d[0];
            s_dist[rb + 1][col] = d[1];
            s_dist[rb + 2][col] = d[2];
            s_dist[rb + 3][col] = d[3];
            s_dist[rb + 4][col] = d[4];
            s_dist[rb + 5][col] = d[5];
            s_dist[rb + 6][col] = d[6];
            s_dist[rb + 7][col] = d[7];
            __syncthreads();

            // ---- per-lane running top-9 for GT == wave ----
#pragma unroll
            for (int t = 0; t < 8; ++t) {
                int c2   = lane + t * 32;
                int gidx = blk * 256 + c2;
                if (gidx < Pl) {
                    float val = s_dist[wave][c2];
                    if (val < tv[N_CAND - 1]) {
                        int pos = N_CAND - 1;
                        while (pos > 0 && tv[pos - 1] > val) {
                            tv[pos] = tv[pos - 1]; tix[pos] = tix[pos - 1]; --pos;
                        }
                        tv[pos] = val; tix[pos] = gidx;
                    }
                }
            }
            __syncthreads();
        }

        // ---- merge 32 lanes x 9 -> global top-9 for this GT/level ----
#pragma unroll
        for (int j = 0; j < N_CAND; ++j) {
            s_mval[wave][lane * N_CAND + j] = tv[j];
            s_midx[wave][lane * N_CAND + j] = tix[j];
        }
        __syncthreads();
        if (lane == 0) {
            for (int r = 0; r < N_CAND; ++r) {
                float best = 3.9e38f; int bslot = 0;
                for (int t = 0; t < N_CAND * 32; ++t) {
                    float v = s_mval[wave][t];
                    if (v < best) { best = v; bslot = t; }
                }
                s_tidx[wave][l * N_CAND + r] = s_midx[wave][bslot];
                s_mval[wave][bslot] = 3.9e38f;
            }
        }
        __syncthreads();

        // ---- pos_ov = IoU(gt, candidate prior box) ----
        if (lane < N_CAND) {
            int pi = s_tidx[wave][l * N_CAND + lane];
            int gp = base + pi;
            float cx = priors[gp * 4 + 0], cy = priors[gp * 4 + 1], s = priors[gp * 4 + 2];
            float pb[4] = { cx - s * 0.5f, cy - s * 0.5f, cx + s * 0.5f, cy + s * 0.5f };
            float gb[4] = { s_box[wave][0], s_box[wave][1], s_box[wave][2], s_box[wave][3] };
            s_posov[wave][l * N_CAND + lane] = iou_xy(gb, pb);
        }
        __syncthreads();
    }

    // ---- per-GT threshold = mean + std (ddof=1) over 45 candidates ----
    if (tid < N_K) {
        float sum = 0.0f, sq = 0.0f;
        for (int t = 0; t < N_LEV * N_CAND; ++t) {
            float v = s_posov[tid][t]; sum += v; sq += v * v;
        }
        float mean = sum / 45.0f;
        float var  = (sq - 45.0f * mean * mean) / 44.0f;
        s_thr[tid] = mean + sqrtf(fmaxf(var, 0.0f));
    }
    __syncthreads();

    // ---- per (level, candidate-slot) argmax over GTs, CIoU, focal corr ----
    if (tid < N_LEV * N_CAND) {
        int l = tid / N_CAND, j = tid % N_CAND;
        int base = 0;
        { const int SPL2[6] = {0, 6400, 8000, 8400, 8500, 8525}; base = SPL2[l]; }
        float best = 0.0f; int bk = 0;
        for (int k = 0; k < N_K; ++k) {
            float ov = s_posov[k][tid];
            int pi = s_tidx[k][tid];
            int gp = base + pi;
            float cx = priors[gp * 4 + 0], cy = priors[gp * 4 + 1];
            bool inside = (s_box[k][0] < cx) && (cx < s_box[k][2]) &&
                          (s_box[k][1] < cy) && (cy < s_box[k][3]);
            float mval = (ov > s_thr[k] && inside) ? ov : 0.0f;
            if (mval > best) { best = mval; bk = k; }
        }
        bool valid = best > 0.0f;
        int  lab   = valid ? s_lab[bk] : 0;

        // decode matched prior, CIoU vs matched GT
        float locnum = 0.0f, locden = 0.0f;
        {
            int pi = s_tidx[bk][tid];
            int gp = base + pi;
            const float* g = &locs[(size_t)(b * N_P + gp) * 4];
            float pcx = priors[gp * 4 + 0], pcy = priors[gp * 4 + 1], ps = priors[gp * 4 + 2];
            float dcx = g[0] * ps * 0.1f + pcx;
            float dcy = g[1] * ps * 0.1f + pcy;
            float dw  = expf(g[2] * 0.2f) * ps;
            float dh  = expf(g[3] * 0.2f) * ps;
            float pb[4] = { dcx - dw * 0.5f, dcy - dh * 0.5f, dcx + dw * 0.5f, dcy + dh * 0.5f };
            float gb[4] = { s_box[bk][0], s_box[bk][1], s_box[bk][2], s_box[bk][3] };
            if (valid) { locnum = 1.0f - ciou_f(pb, gb); locden = 1.0f; }
        }

        // focal correction at the quirk position (prior index SPL[l]+j), class lab-1
        float corr = 0.0f, np = 0.0f;
        if (lab > 0) {
            np = 1.0f;
            float x = scores[(size_t)(b * N_P + base + j) * N_C + (lab - 1)];
            float p  = 1.0f / (1.0f + expf(-x));
            float t1 = (1.0f - p) * (1.0f - p) * logsig(x);
            float t2 = p * p * logsig(-x);
            corr = -F_ALPHA_ * t1 + (1.0f - F_ALPHA_) * t2;
        }
        s_acc[tid][0] = locnum; s_acc[tid][1] = locden;
        s_acc[tid][2] = corr;   s_acc[tid][3] = np;
    }
    __syncthreads();
    if (tid == 0) {
        float a0 = 0, a1 = 0, a2 = 0, a3 = 0;
        for (int t = 0; t < N_LEV * N_CAND; ++t) {
            a0 += s_acc[t][0]; a1 += s_acc[t][1]; a2 += s_acc[t][2]; a3 += s_acc[t][3];
        }
        img_part[b * 4 + 0] = a0; img_part[b * 4 + 1] = a1;
        img_part[b * 4 + 2] = a2; img_part[b * 4 + 3] = a3;
    }
}

// ------------------------------------------------------------------
// Kernel B: background focal sum over all 21.8M logits (t=0 everywhere);
// streaming float4 reduction, double accumulation, deterministic partials.
// ------------------------------------------------------------------
__device__ __forceinline__ double bg_term(float x) {
    float p = 1.0f / (1.0f + expf(-x));
    return (double)(-(1.0f - F_ALPHA_) * (p * p) * logsig(-x));
}

__global__ void __launch_bounds__(256)
focal_bg_kernel(const float4* __restrict__ x4, int n4, double* __restrict__ part)
{
    __shared__ double red[256];
    const int tid    = threadIdx.x;
    const int stride = gridDim.x * blockDim.x;
    double acc = 0.0;
    for (int i = blockIdx.x * blockDim.x + tid; i < n4; i += stride) {
        __builtin_prefetch(&x4[i + stride], 0, 1);   // global_prefetch_b8 (speculative)
        float4 v = x4[i];
        acc += bg_term(v.x) + bg_term(v.y) + bg_term(v.z) + bg_term(v.w);
    }
    red[tid] = acc;
    __syncthreads();
    for (int s = 128; s > 0; s >>= 1) {
        if (tid < s) red[tid] += red[tid + s];
        __syncthreads();
    }
    if (tid == 0) part[blockIdx.x] = red[0];
}

// ------------------------------------------------------------------
// Kernel C: deterministic final combine.
// ------------------------------------------------------------------
__global__ void finalize_kernel(const double* __restrict__ part, int nblk,
                                const float* __restrict__ img_part,
                                float* __restrict__ out)
{
    if (threadIdx.x == 0 && blockIdx.x == 0) {
        double bg = 0.0;
        for (int i = 0; i < nblk; ++i) bg += part[i];
        float locnum = 0, locden = 0, corr = 0, np = 0;
        for (int bimg = 0; bimg < N_B; ++bimg) {
            locnum += img_part[bimg * 4 + 0];
            locden += img_part[bimg * 4 + 1];
            corr   += img_part[bimg * 4 + 2];
            np     += img_part[bimg * 4 + 3];
        }
        float conf = (float)((bg + (double)corr) / (double)np);
        float loc  = locnum / fmaxf(locden, 1.0f);
        out[0] = conf + loc;
    }
}

extern "C" void kernel_launch(void* const* d_in, const int* in_sizes, int n_in,
                              void* d_out, int out_size, void* d_ws, size_t ws_size,
                              hipStream_t stream)
{
    const float* locs   = (const float*)d_in[0];   // (B,P,4)
    const float* scores = (const float*)d_in[1];   // (B,P,C)
    const float* boxes  = (const float*)d_in[2];   // (B,K,4)
    const int*   labels = (const int*)d_in[3];     // (B,K)
    const float* priors = (const float*)d_in[4];   // (P,4)
    float* out = (float*)d_out;

    double* part     = (double*)d_ws;                                   // NBLK_BG doubles
    float*  img_part = (float*)((char*)d_ws + NBLK_BG * sizeof(double)); // 32*4 floats

    const int n4 = (N_B * N_P * N_C) / 4;   // 5,456,000 float4 elements

    focal_bg_kernel<<<NBLK_BG, 256, 0, stream>>>((const float4*)scores, n4, part);
    atss_assign_kernel<<<N_B, 512, 0, stream>>>(locs, scores, boxes, labels, priors, img_part);
    finalize_kernel<<<1, 64, 0, stream>>>(part, NBLK_BG, img_part, out);
}